// MPSWavefunction_67697274520167
// MI455X (gfx1250) — compile-verified
//
#include <hip/hip_runtime.h>
#include <stdint.h>

#define DMAX    128
#define NSITES  48
#define THREADS 256
#define MELEMS  (DMAX * DMAX)   // 16384 floats = 64 KB per matrix buffer

typedef float v2f __attribute__((ext_vector_type(2)));
typedef float v8f __attribute__((ext_vector_type(8)));

__device__ __forceinline__ uint32_t lds_addr32(const void* p) {
  // Generic pointer to LDS: low 32 bits are the wave-relative LDS byte address.
  return (uint32_t)(uintptr_t)p;
}

// Async Global -> LDS copy of sz floats (sz uniform across the workgroup).
// 16B chunks per lane via global_load_async_to_lds_b128, 4B tail via _b32.
// Tracked on ASYNCcnt; caller must s_wait_asynccnt + barrier before reading LDS.
__device__ __forceinline__ void async_copy_matrix(const float* __restrict__ gsrc,
                                                  float* lds_dst, int sz, int tid) {
  const int n128 = sz >> 2;           // number of 16-byte chunks
  for (int i = tid; i < n128; i += THREADS) {
    uint32_t la = lds_addr32(lds_dst + 4 * i);
    uint64_t ga = (uint64_t)(uintptr_t)(gsrc + 4 * i);
    asm volatile("global_load_async_to_lds_b128 %0, %1, off"
                 :: "v"(la), "v"(ga) : "memory");
  }
  const int rem = sz & 3;             // 0..3 trailing dwords
  if (tid < rem) {
    uint32_t la = lds_addr32(lds_dst + 4 * n128 + tid);
    uint64_t ga = (uint64_t)(uintptr_t)(gsrc + 4 * n128 + tid);
    asm volatile("global_load_async_to_lds_b32 %0, %1, off"
                 :: "v"(la), "v"(ga) : "memory");
  }
}

__device__ __forceinline__ void wait_async_all() {
#if __has_builtin(__builtin_amdgcn_s_wait_asynccnt)
  __builtin_amdgcn_s_wait_asynccnt(0);
#else
  asm volatile("s_wait_asynccnt 0x0" ::: "memory");
#endif
}

__global__ __launch_bounds__(THREADS)
void mps_chain_kernel(const float* __restrict__ data,
                      const float* __restrict__ sgn,
                      const int* __restrict__ qidx,          // [NB][NSITES][3] (istat, dr, dc)
                      const unsigned char* __restrict__ symb,
                      float* __restrict__ out)
{
  __shared__ float Mbuf[2][MELEMS];   // double-buffered matrix slabs
  __shared__ float Vbuf[2][DMAX];     // ping-pong vectors

  const int b    = blockIdx.x;
  const int tid  = threadIdx.x;
  const int lane = tid & 31;
  const int w    = tid >> 5;          // wave id 0..7 -> 16-row stripe
  const int kh   = lane >> 4;         // K-half (lanes 0-15 vs 16-31)
  const int row  = (w << 4) + (lane & 15);

  const int* qb = qidx + b * NSITES * 3;

  // x = e0 (length-1 vector since bonds[-1] == 1)
  if (tid < DMAX) {
    Vbuf[0][tid] = (tid == 0) ? 1.0f : 0.0f;
    Vbuf[1][tid] = 0.0f;
  }

  // Prefetch last site's matrix into Mbuf[0] + zero guard band after it.
  {
    const int* q  = qb + (NSITES - 1) * 3;
    const int  sz = q[1] * q[2];
    async_copy_matrix(data + q[0], Mbuf[0], sz, tid);
    const int zend = min(sz + 512, MELEMS);
    for (int i = sz + tid; i < zend; i += THREADS) Mbuf[0][i] = 0.0f;
  }
  wait_async_all();
  __syncthreads();

  int cur = 0, xv = 0;
  for (int s = NSITES - 1; s >= 0; --s) {
    const int dr = qb[s * 3 + 1];
    const int dc = qb[s * 3 + 2];

    // Overlap: start streaming the next site's matrix into the other buffer.
    if (s > 0) {
      const int* qn = qb + (s - 1) * 3;
      const int  sz = qn[1] * qn[2];
      async_copy_matrix(data + qn[0], Mbuf[cur ^ 1], sz, tid);
      const int zend = min(sz + 512, MELEMS);
      for (int i = sz + tid; i < zend; i += THREADS) Mbuf[cur ^ 1][i] = 0.0f;
    }

    // y = M x with V_WMMA_F32_16X16X4_F32, broadcast-column trick:
    // every column of B is the current x chunk, so every column of D is y.
    const float* Mc = Mbuf[cur];   // column-major dr x dc, tight stride dr
    const float* x  = Vbuf[xv];
    float*       y  = Vbuf[xv ^ 1];

    v8f acc = {};
    const int kbmax = (dc + 3) >> 2;
    for (int kb = 0; kb < kbmax; ++kb) {
      const int c0 = (kb << 2) + (kh << 1);   // this lane's K pair
      v2f a, bm;
      a[0]  = Mc[c0 * dr + row];
      a[1]  = Mc[(c0 + 1) * dr + row];
      bm[0] = x[c0];
      bm[1] = x[c0 + 1];
      acc = __builtin_amdgcn_wmma_f32_16x16x4_f32(
          /*neg_a=*/false, a, /*neg_b=*/false, bm,
          /*c_mod=*/(short)0, acc, /*reuse_a=*/false, /*reuse_b=*/false);
    }

    // Column N=0 of D: lane 0 holds rows 16w..16w+7, lane 16 rows 16w+8..16w+15.
    if ((lane & 15) == 0) {
      const int rbase = (w << 4) + (kh << 3);
      #pragma unroll
      for (int v = 0; v < 8; ++v) y[rbase + v] = acc[v];
    }
    __syncthreads();
    if (tid < DMAX && tid >= dr) y[tid] = 0.0f;   // keep x zero past logical length
    wait_async_all();                              // next matrix fully in LDS
    __syncthreads();

    cur ^= 1;
    xv  ^= 1;
  }

  if (tid == 0) {
    const float a = Vbuf[xv][0];
    out[b] = (symb[b] ? 0.0f : a) * sgn[b];
  }
}

extern "C" void kernel_launch(void* const* d_in, const int* in_sizes, int n_in,
                              void* d_out, int out_size, void* d_ws, size_t ws_size,
                              hipStream_t stream) {
  const float*         data = (const float*)d_in[0];
  const float*         sgn  = (const float*)d_in[1];
  const int*           qidx = (const int*)d_in[2];          // int32 after JAX downcast
  const unsigned char* symb = (const unsigned char*)d_in[3]; // bool -> 1 byte
  float* out = (float*)d_out;

  dim3 grid((unsigned)out_size);   // 512 samples, one workgroup each
  dim3 block(THREADS);
  hipLaunchKernelGGL(mps_chain_kernel, grid, block, 0, stream,
                     data, sgn, qidx, symb, out);
}